// Attn_43843026157961
// MI455X (gfx1250) — compile-verified
//
#include <hip/hip_runtime.h>
#include <hip/hip_bf16.h>

// Problem sizes (fixed by the reference).
#define BB 64
#define SS 2048
#define HH 512
#define W2COLS (2 * HH)   // W_attn row stride

typedef __attribute__((ext_vector_type(2))) float v2f;
typedef __attribute__((ext_vector_type(4))) float v4f;
typedef __attribute__((ext_vector_type(8))) float v8f;

// ---------------------------------------------------------------------------
// Phase 0:  u[k] = sum_h v[h] * W_attn[h, HH + k]      (u = W2^T v, 512 floats)
// 8 blocks x 256 threads; block owns a 64-wide k-chunk, 4 h-slices per block,
// LDS combine. Total traffic ~1 MB — negligible next to the 256 MiB main pass.
// ---------------------------------------------------------------------------
__global__ __launch_bounds__(256) void compute_u_kernel(
    const float* __restrict__ W, const float* __restrict__ v,
    float* __restrict__ u) {
  __shared__ float part[4][64];
  const int t  = threadIdx.x;
  const int kl = t & 63;                 // k within chunk
  const int hs = t >> 6;                 // h-slice 0..3 (128 h each)
  const int k  = blockIdx.x * 64 + kl;

  const float* col = W + (size_t)(hs * 128) * W2COLS + HH + k;
  float acc = 0.0f;
  #pragma unroll 4
  for (int h = 0; h < 128; ++h) {
    acc += v[hs * 128 + h] * col[(size_t)h * W2COLS];
  }
  part[hs][kl] = acc;
  __syncthreads();
  if (hs == 0) {
    u[k] = part[0][kl] + part[1][kl] + part[2][kl] + part[3][kl];
  }
}

// ---------------------------------------------------------------------------
// Phase 1:  scores[b,s] = enc[b,s,:] . u   via V_WMMA_F32_16X16X4_F32.
//
// One wave owns 16 consecutive s-rows of one b. Loop K in steps of 4:
//   A (16x4 f32, documented layout): lane l holds row M=(l&15),
//       K = vgpr + 2*(l>>4)  ->  per lane a b64 load of enc[row, kb+2*(l>>4)].
//   B (4x16 f32): replicate u across all N columns; with the half-splits-K
//       convention, lane l needs {u[kb+2h], u[kb+1+2h]}, h = l>>4  ->  one
//       ds_load_b64 from u staged in LDS.
//   C/D (16x16 f32): every column N equals score(s_base+M); VGPR r holds
//       M=r (lanes 0-15) and M=r+8 (lanes 16-31). Lanes 0 and 16 store 8
//       floats each as two b128 stores.
//
// enc is read exactly once -> non-temporal loads to avoid polluting L2.
// ---------------------------------------------------------------------------
__global__ __launch_bounds__(128) void attn_scores_kernel(
    const float* __restrict__ enc, const float* __restrict__ u,
    float* __restrict__ out) {
  __shared__ float lds_u[HH];
  const int t = threadIdx.x;
  // Stage u (2 KB) into LDS, 4 floats per thread.
  #pragma unroll
  for (int i = 0; i < HH / 128; ++i) lds_u[t + i * 128] = u[t + i * 128];
  __syncthreads();

  const int lane  = t & 31;
  const int wave  = t >> 5;                        // 0..3
  const int gwave = blockIdx.x * 4 + wave;         // 0..8191
  const int b     = gwave >> 7;                    // / (SS/16)
  const int stile = gwave & 127;                   // 16-row tile within S
  const int half  = lane >> 4;                     // 0: K+{0,1}, 1: K+{2,3}
  const int row   = (stile << 4) + (lane & 15);    // s index this lane feeds

  const float* arow = enc + ((size_t)b * SS + row) * HH + 2 * half;
  const float* brow = lds_u + 2 * half;

  v8f c = {};
  #pragma unroll 8
  for (int kb = 0; kb < HH; kb += 4) {
    v2f a  = __builtin_nontemporal_load((const v2f*)(arow + kb)); // use-once stream
    v2f bv = *(const v2f*)(brow + kb);                            // ds_load_b64
    // 8 args: (neg_a, A, neg_b, B, c_mod, C, reuse_a, reuse_b)
    c = __builtin_amdgcn_wmma_f32_16x16x4_f32(
        false, a, false, bv, (short)0, c, false, false);
  }

  // Column 0 of each lane-half carries the 16 scores of this tile.
  if ((lane & 15) == 0) {
    float* o = out + (size_t)b * SS + (stile << 4) + (half << 3);
    v4f lo = {c[0], c[1], c[2], c[3]};
    v4f hi = {c[4], c[5], c[6], c[7]};
    *(v4f*)(o)     = lo;
    *(v4f*)(o + 4) = hi;
  }
}

// ---------------------------------------------------------------------------
// Phase 2: in-place softmax over S for each b row (scores sit in L2: 512 KB).
// One 256-thread block per row; 8 values per thread; LDS tree reductions.
// ---------------------------------------------------------------------------
__global__ __launch_bounds__(256) void softmax_rows_kernel(float* __restrict__ out) {
  __shared__ float red[256];
  const int t = threadIdx.x;
  float* row = out + (size_t)blockIdx.x * SS;

  float vals[8];
  float m = -3.402823466e38f;
  #pragma unroll
  for (int i = 0; i < 8; ++i) {
    vals[i] = row[t + i * 256];
    m = fmaxf(m, vals[i]);
  }
  red[t] = m;
  __syncthreads();
  for (int off = 128; off > 0; off >>= 1) {
    if (t < off) red[t] = fmaxf(red[t], red[t + off]);
    __syncthreads();
  }
  m = red[0];
  __syncthreads();

  float s = 0.0f;
  #pragma unroll
  for (int i = 0; i < 8; ++i) {
    vals[i] = __expf(vals[i] - m);
    s += vals[i];
  }
  red[t] = s;
  __syncthreads();
  for (int off = 128; off > 0; off >>= 1) {
    if (t < off) red[t] += red[t + off];
    __syncthreads();
  }
  const float inv = 1.0f / red[0];
  #pragma unroll
  for (int i = 0; i < 8; ++i) row[t + i * 256] = vals[i] * inv;
}

// ---------------------------------------------------------------------------
// Launch: inputs are (hidden, encoder_outputs, W_attn, b_attn, v).
// hidden and b_attn contribute only per-row constants, which softmax over S
// annihilates exactly — they are intentionally unused.
// ---------------------------------------------------------------------------
extern "C" void kernel_launch(void* const* d_in, const int* in_sizes, int n_in,
                              void* d_out, int out_size, void* d_ws, size_t ws_size,
                              hipStream_t stream) {
  (void)in_sizes; (void)n_in; (void)out_size; (void)ws_size;
  const float* enc = (const float*)d_in[1];   // [B, S, H]
  const float* W   = (const float*)d_in[2];   // [H, 2H]
  const float* v   = (const float*)d_in[4];   // [H]
  float* out = (float*)d_out;                 // [B, S]
  float* u   = (float*)d_ws;                  // [H] scratch (2 KB)

  compute_u_kernel<<<HH / 64, 256, 0, stream>>>(W, v, u);
  attn_scores_kernel<<<(BB * SS / 16) / 4, 128, 0, stream>>>(enc, u, out);
  softmax_rows_kernel<<<BB, 256, 0, stream>>>(out);
}